// MSGCN_19894288515310
// MI455X (gfx1250) — compile-verified
//
#include <hip/hip_runtime.h>
#include <hip/hip_bf16.h>
#include <math.h>

// ---------------------------------------------------------------------------
// Problem constants (from reference): B=16, D=256, N=64, NB=5, heads=4, dh=64
// ---------------------------------------------------------------------------
#define BB   16
#define DD   256
#define NN   64
#define NBANDS 5

typedef __attribute__((ext_vector_type(2))) float v2f;
typedef __attribute__((ext_vector_type(8))) float v8f;

// ---------------------------------------------------------------------------
// Generic strided / batched f32 WMMA GEMM:  D = A(MxK) * B(KxN) [+bias][gelu]
// One wave computes one 16x16 output tile using V_WMMA_F32_16X16X4_F32.
// A[m][k] = A[m*rsA + k*csA]  (strides allow implicit transposes)
// Template flags: AC -> csA==1 (A K-contiguous, use b64 loads)
//                 BC -> rsB==1 (B K-contiguous, use b64 loads)
// batch z = outer*innerCount + inner; per-axis base offsets for A/B/D/bias.
// epi: 0 = none, 1 = +bias(col), 2 = gelu(x + bias(col))  (exact erf GELU)
// ---------------------------------------------------------------------------
template <bool AC, bool BC>
__global__ void __launch_bounds__(256)
wmma_gemm_f32(const float* __restrict__ A, const float* __restrict__ Bm,
              const float* __restrict__ bias, float* __restrict__ Dm,
              int M, int N, int K,
              int rsA, int csA, int rsB, int csB, int rsD, int csD,
              long aO, long aI, long bO, long bI, long dO, long dI, long biasO,
              int innerCount, int epi)
{
    int z     = blockIdx.z;
    int outer = z / innerCount;
    int inner = z % innerCount;
    A  += outer * aO + inner * aI;
    Bm += outer * bO + inner * bI;
    Dm += outer * dO + inner * dI;
    if (bias) bias += outer * biasO;

    int wave = threadIdx.x >> 5;
    int lane = threadIdx.x & 31;
    int tilesN = N >> 4;
    int tilesM = M >> 4;
    int tile = blockIdx.x * 8 + wave;     // 8 waves per 256-thread block
    if (tile >= tilesM * tilesN) return;  // wave-uniform: EXEC stays all-ones

    int mo = (tile / tilesN) << 4;
    int no = (tile % tilesN) << 4;
    int half = lane >> 4;                 // K pair select (ISA 32-bit A layout)
    int mr   = lane & 15;                 // A row within tile / B,C,D column

    v8f acc = {};
    const float* Ab = A  + (long)(mo + mr) * rsA;
    const float* Bb = Bm + (long)(no + mr) * csB;

    #pragma unroll 2
    for (int k0 = 0; k0 < K; k0 += 4) {
        int ka = k0 + 2 * half;
        v2f av, bv;
        if (AC) {                                   // csA == 1, ka even -> 8B aligned
            av = *(const v2f*)(Ab + ka);
        } else {
            av.x = Ab[(long)ka       * csA];
            av.y = Ab[(long)(ka + 1) * csA];
        }
        if (BC) {                                   // rsB == 1
            bv = *(const v2f*)(Bb + ka);
        } else {
            bv.x = Bb[(long)ka       * rsB];
            bv.y = Bb[(long)(ka + 1) * rsB];
        }
        acc = __builtin_amdgcn_wmma_f32_16x16x4_f32(
                  false, av, false, bv, (short)0, acc, false, false);
    }

    int n = no + mr;
    float bvad = (epi >= 1 && bias) ? bias[n] : 0.0f;
    #pragma unroll
    for (int r = 0; r < 8; ++r) {
        int m = mo + r + 8 * half;        // ISA C/D layout: halves split M 0-7/8-15
        float v = acc[r] + bvad;
        if (epi == 2) v = 0.5f * v * (1.0f + erff(v * 0.70710678118654752f));
        Dm[(long)m * rsD + (long)n * csD] = v;
    }
}

// ---------------------------------------------------------------------------
// staticpos[band,i,j] = softmax(adj_w[band,i,:])[j] + pf[band]*softmax(pos)[j]
// ---------------------------------------------------------------------------
__global__ void static_pos_kernel(const float* __restrict__ adj_w,
                                  const float* __restrict__ pos_enc,
                                  const float* __restrict__ pf,
                                  float* __restrict__ sp)
{
    int band = blockIdx.x / NN, i = blockIdx.x % NN, tid = threadIdx.x;
    __shared__ float rowA[NN], rowP[NN], sc[4];
    long base = ((long)band * NN + i) * NN;
    rowA[tid] = adj_w[base + tid];
    rowP[tid] = pos_enc[base + tid];
    __syncthreads();
    if (tid == 0) {
        float ma = -1e30f, mp = -1e30f;
        for (int j = 0; j < NN; ++j) { ma = fmaxf(ma, rowA[j]); mp = fmaxf(mp, rowP[j]); }
        float da = 0.f, dp = 0.f;
        for (int j = 0; j < NN; ++j) { da += expf(rowA[j] - ma); dp += expf(rowP[j] - mp); }
        sc[0] = ma; sc[1] = da; sc[2] = mp; sc[3] = dp;
    }
    __syncthreads();
    float v = expf(rowA[tid] - sc[0]) / sc[1]
            + pf[band] * expf(rowP[tid] - sc[2]) / sc[3];
    sp[base + tid] = v;
}

// ---------------------------------------------------------------------------
// Fused edge-MLP + row softmax + degree norm.  Block = (band,b,i), 256 thr.
// Wave-parallel: wave w owns rows j = 8w..8w+7; lane owns 8 hidden channels.
// dyn_j = sigmoid( sum_h elu(a[i,h]+bp[j,h]+sb1[h]) * w2[h] + sb2 )
// adj   = softmax_j(staticpos + dyn) * 1/(1+1e-8)   (deg == 1 after softmax)
// Never materializes the 335MB h tensor; no __syncthreads in the main loop.
// ---------------------------------------------------------------------------
__global__ void __launch_bounds__(256)
local_dyn_adj_kernel(const float* __restrict__ a_buf, const float* __restrict__ bp_buf,
                     const float* __restrict__ sb1, const float* __restrict__ sw2,
                     const float* __restrict__ sb2, const float* __restrict__ spbuf,
                     float* __restrict__ adj_out)
{
    int band = blockIdx.x / (BB * NN);
    int rem  = blockIdx.x % (BB * NN);
    int b = rem >> 6, i = rem & 63;
    int tid = threadIdx.x;
    int wave = tid >> 5, lane = tid & 31;

    __shared__ float dynr[NN];
    __shared__ float sm[2];

    long rowbase = (((long)(band * BB + b) * NN) + i) * DD;
    long bpbase  = ((long)(band * BB + b) * NN) * DD;

    float av[8], s1[8], w2v[8];
    #pragma unroll
    for (int u = 0; u < 8; ++u) {
        int h = lane + 32 * u;                    // coalesced within wave
        av[u]  = a_buf[rowbase + h];
        s1[u]  = sb1[band * DD + h];
        w2v[u] = sw2[band * DD + h];
    }
    float b2 = sb2[band];

    #pragma unroll
    for (int jj = 0; jj < 8; ++jj) {
        int j = wave * 8 + jj;
        const float* bpr = bp_buf + bpbase + (long)j * DD;
        float p = 0.f;
        #pragma unroll
        for (int u = 0; u < 8; ++u) {
            float t = av[u] + bpr[lane + 32 * u] + s1[u];
            float e = (t > 0.0f) ? t : expm1f(t); // exact ELU (alpha=1)
            p += e * w2v[u];
        }
        #pragma unroll
        for (int off = 16; off >= 1; off >>= 1) p += __shfl_xor(p, off, 32);
        if (lane == 0) dynr[j] = 1.0f / (1.0f + expf(-(p + b2)));
    }
    __syncthreads();

    if (tid < NN) dynr[tid] += spbuf[((long)(band * NN + i)) * NN + tid];
    __syncthreads();
    if (tid == 0) {
        float mx = -1e30f;
        for (int j = 0; j < NN; ++j) mx = fmaxf(mx, dynr[j]);
        float s = 0.f;
        for (int j = 0; j < NN; ++j) s += expf(dynr[j] - mx);
        sm[0] = mx; sm[1] = s;
    }
    __syncthreads();
    if (tid < NN) {
        float v = expf(dynr[tid] - sm[0]) / sm[1];
        v *= (1.0f / (1.0f + 1e-8f));             // dis_i * dis_j with deg == 1
        adj_out[(((long)(band * BB + b) * NN) + i) * NN + tid] = v;
    }
}

// scale_nodes[b,band,d] = mean_n local_out[b,band,d,n]
__global__ void mean_over_n_kernel(const float* __restrict__ lo, float* __restrict__ sn)
{
    int t = blockIdx.x * blockDim.x + threadIdx.x;
    if (t >= BB * NBANDS * DD) return;
    const float* p = lo + (long)t * NN;
    float s = 0.f;
    for (int n = 0; n < NN; ++n) s += p[n];
    sn[t] = s * (1.0f / NN);
}

// 4-head attention over S=5 tokens. One wave per head, lane owns 2 channels.
__global__ void __launch_bounds__(128)
attn_core_kernel(const float* __restrict__ qkv, float* __restrict__ o)
{
    int b = blockIdx.x;
    int h = threadIdx.x >> 5;
    int lane = threadIdx.x & 31;
    int d0 = lane * 2;
    const float* base = qkv + (long)b * 5 * 768;
    float q[5][2], k[5][2], v[5][2];
    #pragma unroll
    for (int i = 0; i < 5; ++i) {
        const float* r = base + i * 768 + h * 64 + d0;
        q[i][0] = r[0];       q[i][1] = r[1];
        k[i][0] = r[256];     k[i][1] = r[257];
        v[i][0] = r[512];     v[i][1] = r[513];
    }
    float att[5][5];
    #pragma unroll
    for (int i = 0; i < 5; ++i)
        #pragma unroll
        for (int j = 0; j < 5; ++j) {
            float p = q[i][0] * k[j][0] + q[i][1] * k[j][1];
            for (int off = 16; off >= 1; off >>= 1) p += __shfl_xor(p, off, 32);
            att[i][j] = p * 0.125f;                    // 1/sqrt(dh=64)
        }
    #pragma unroll
    for (int i = 0; i < 5; ++i) {
        float mx = att[i][0];
        for (int j = 1; j < 5; ++j) mx = fmaxf(mx, att[i][j]);
        float s = 0.f;
        for (int j = 0; j < 5; ++j) { att[i][j] = expf(att[i][j] - mx); s += att[i][j]; }
        float inv = 1.0f / s;
        float o0 = 0.f, o1 = 0.f;
        for (int j = 0; j < 5; ++j) { o0 += att[i][j] * v[j][0]; o1 += att[i][j] * v[j][1]; }
        o[((long)b * 5 + i) * DD + h * 64 + d0    ] = o0 * inv;
        o[((long)b * 5 + i) * DD + h * 64 + d0 + 1] = o1 * inv;
    }
}

// out = LayerNorm(x + y), row length 256, one block per row
__global__ void __launch_bounds__(256)
ln_residual_kernel(const float* __restrict__ x, const float* __restrict__ y,
                   const float* __restrict__ g, const float* __restrict__ be,
                   float* __restrict__ out)
{
    int row = blockIdx.x, tid = threadIdx.x;
    __shared__ float red[256];
    float v = x[(long)row * DD + tid] + y[(long)row * DD + tid];
    red[tid] = v; __syncthreads();
    for (int s = 128; s > 0; s >>= 1) { if (tid < s) red[tid] += red[tid + s]; __syncthreads(); }
    float m = red[0] * (1.0f / DD);
    __syncthreads();
    float dv = v - m;
    red[tid] = dv * dv; __syncthreads();
    for (int s = 128; s > 0; s >>= 1) { if (tid < s) red[tid] += red[tid + s]; __syncthreads(); }
    float var = red[0] * (1.0f / DD);
    float r = rsqrtf(var + 1e-5f);
    out[(long)row * DD + tid] = dv * r * g[tid] + be[tid];
}

// Global 5-node graph: fused edge MLP + softmax + degree norm -> g_adj output
// Waves 0..4 each own one j; lane owns 8 hidden channels.
__global__ void __launch_bounds__(256)
g_dyn_adj_kernel(const float* __restrict__ ag, const float* __restrict__ bg,
                 const float* __restrict__ gsb1, const float* __restrict__ gsw2,
                 const float* __restrict__ gsb2, const float* __restrict__ gadj_w,
                 const float* __restrict__ gpos, const float* __restrict__ gpf,
                 float* __restrict__ gadj_out)
{
    int b = blockIdx.x / NBANDS, i = blockIdx.x % NBANDS, tid = threadIdx.x;
    int wave = tid >> 5, lane = tid & 31;
    __shared__ float dynr[NBANDS];

    if (wave < NBANDS) {
        int j = wave;
        long abase = ((long)b * NBANDS + i) * DD;
        long bbase = ((long)b * NBANDS + j) * DD;
        float p = 0.f;
        #pragma unroll
        for (int u = 0; u < 8; ++u) {
            int h = lane + 32 * u;
            float t = ag[abase + h] + bg[bbase + h] + gsb1[h];
            float e = (t > 0.0f) ? t : expm1f(t);
            p += e * gsw2[h];
        }
        #pragma unroll
        for (int off = 16; off >= 1; off >>= 1) p += __shfl_xor(p, off, 32);
        if (lane == 0) dynr[j] = 1.0f / (1.0f + expf(-(p + gsb2[0])));
    }
    __syncthreads();

    if (tid == 0) {
        float sa[NBANDS], sp[NBANDS], comb[NBANDS];
        float ma = -1e30f, mp = -1e30f;
        for (int j = 0; j < NBANDS; ++j) { ma = fmaxf(ma, gadj_w[i*NBANDS+j]); mp = fmaxf(mp, gpos[i*NBANDS+j]); }
        float da = 0.f, dp = 0.f;
        for (int j = 0; j < NBANDS; ++j) {
            sa[j] = expf(gadj_w[i*NBANDS+j] - ma); da += sa[j];
            sp[j] = expf(gpos[i*NBANDS+j] - mp);   dp += sp[j];
        }
        float mx = -1e30f;
        for (int j = 0; j < NBANDS; ++j) { comb[j] = sa[j]/da + gpf[0]*sp[j]/dp + dynr[j]; mx = fmaxf(mx, comb[j]); }
        float ds = 0.f;
        for (int j = 0; j < NBANDS; ++j) { comb[j] = expf(comb[j] - mx); ds += comb[j]; }
        float sc = (1.0f / (1.0f + 1e-8f)) / ds;
        for (int j = 0; j < NBANDS; ++j) gadj_out[(long)b*25 + i*NBANDS + j] = comb[j] * sc;
    }
}

// tmp_g[b,i,d] = sum_j g_adj[b,i,j] * se[b,j,d]
__global__ void global_agg_kernel(const float* __restrict__ gadj,
                                  const float* __restrict__ se, float* __restrict__ tg)
{
    int t = blockIdx.x * blockDim.x + threadIdx.x;
    if (t >= BB * NBANDS * DD) return;
    int d = t & (DD - 1);
    int bi = t >> 8;
    int b = bi / NBANDS, i = bi % NBANDS;
    float s = 0.f;
    for (int j = 0; j < NBANDS; ++j)
        s += gadj[(long)b*25 + i*NBANDS + j] * se[((long)b*NBANDS + j) * DD + d];
    tg[t] = s;
}

// out[b,band,d] = g_out[b,band,d] * (sum_n lo[b,band,d,n]*imp_w[band,n]) + imp_b[band]
__global__ void final_proj_kernel(const float* __restrict__ lo, const float* __restrict__ gout,
                                  const float* __restrict__ imp_w, const float* __restrict__ imp_b,
                                  float* __restrict__ out)
{
    int t = blockIdx.x * blockDim.x + threadIdx.x;
    if (t >= BB * NBANDS * DD) return;
    int band = (t >> 8) % NBANDS;
    const float* p = lo + (long)t * NN;
    const float* w = imp_w + band * NN;
    float s = 0.f;
    for (int n = 0; n < NN; ++n) s += p[n] * w[n];
    out[t] = gout[t] * s + imp_b[band];
}

// ---------------------------------------------------------------------------
static inline void launch_gemm(hipStream_t st, const float* A, const float* Bm,
                               const float* bias, float* Dm, int M, int N, int K,
                               int rsA, int csA, int rsB, int csB, int rsD, int csD,
                               long aO, long aI, long bO, long bI, long dO, long dI,
                               long biasO, int innerCount, int outerCount, int epi)
{
    int tiles = (M / 16) * (N / 16);
    dim3 grid((tiles + 7) / 8, 1, innerCount * outerCount);
    bool ac = (csA == 1), bc = (rsB == 1);
    if (ac && bc)
        hipLaunchKernelGGL((wmma_gemm_f32<true, true>), grid, dim3(256), 0, st,
                           A, Bm, bias, Dm, M, N, K, rsA, csA, rsB, csB, rsD, csD,
                           aO, aI, bO, bI, dO, dI, biasO, innerCount, epi);
    else if (ac && !bc)
        hipLaunchKernelGGL((wmma_gemm_f32<true, false>), grid, dim3(256), 0, st,
                           A, Bm, bias, Dm, M, N, K, rsA, csA, rsB, csB, rsD, csD,
                           aO, aI, bO, bI, dO, dI, biasO, innerCount, epi);
    else if (!ac && bc)
        hipLaunchKernelGGL((wmma_gemm_f32<false, true>), grid, dim3(256), 0, st,
                           A, Bm, bias, Dm, M, N, K, rsA, csA, rsB, csB, rsD, csD,
                           aO, aI, bO, bI, dO, dI, biasO, innerCount, epi);
    else
        hipLaunchKernelGGL((wmma_gemm_f32<false, false>), grid, dim3(256), 0, st,
                           A, Bm, bias, Dm, M, N, K, rsA, csA, rsB, csB, rsD, csD,
                           aO, aI, bO, bI, dO, dI, biasO, innerCount, epi);
}

extern "C" void kernel_launch(void* const* d_in, const int* in_sizes, int n_in,
                              void* d_out, int out_size, void* d_ws, size_t ws_size,
                              hipStream_t stream)
{
    const float* feature  = (const float*)d_in[0];
    const float* l_adj_w  = (const float*)d_in[1];
    const float* l_sim_w1 = (const float*)d_in[2];
    const float* l_sim_b1 = (const float*)d_in[3];
    const float* l_sim_w2 = (const float*)d_in[4];
    const float* l_sim_b2 = (const float*)d_in[5];
    const float* l_pf     = (const float*)d_in[6];
    const float* l_pos    = (const float*)d_in[7];
    const float* l_gcn_w  = (const float*)d_in[8];
    const float* l_gcn_b  = (const float*)d_in[9];
    const float* g_adj_w  = (const float*)d_in[10];
    const float* g_sim_w1 = (const float*)d_in[11];
    const float* g_sim_b1 = (const float*)d_in[12];
    const float* g_sim_w2 = (const float*)d_in[13];
    const float* g_sim_b2 = (const float*)d_in[14];
    const float* g_pf     = (const float*)d_in[15];
    const float* g_pos    = (const float*)d_in[16];
    const float* g_gcn_w  = (const float*)d_in[17];
    const float* g_gcn_b  = (const float*)d_in[18];
    const float* qkv_w    = (const float*)d_in[19];
    const float* qkv_b    = (const float*)d_in[20];
    const float* attn_ow  = (const float*)d_in[21];
    const float* attn_ob  = (const float*)d_in[22];
    const float* ln1_g    = (const float*)d_in[23];
    const float* ln1_b    = (const float*)d_in[24];
    const float* ln2_g    = (const float*)d_in[25];
    const float* ln2_b    = (const float*)d_in[26];
    const float* mlp_w1   = (const float*)d_in[27];
    const float* mlp_b1   = (const float*)d_in[28];
    const float* mlp_w2   = (const float*)d_in[29];
    const float* mlp_b2   = (const float*)d_in[30];
    const float* imp_w    = (const float*)d_in[31];
    const float* imp_b    = (const float*)d_in[32];

    float* out  = (float*)d_out;
    float* out1 = out;                         // [B, 5*D]          20480
    float* gadj_out = out + 20480;             // [B, 5, 5]           400
    float* out3 = out + 20880;                 // local_out [B,5,D,N] 1310720

    float* ws = (float*)d_ws;
    float* a_buf   = ws;              // [5,B,N,D]  1310720
    float* bp_buf  = ws + 1310720;    // [5,B,N,D]  1310720
    float* adj_buf = ws + 2621440;    // [5,B,N,N]   327680
    float* spbuf   = ws + 2949120;    // [5,N,N]      20480
    float* tmp_gcn = ws + 2969600;    // [5,B,N,D]  1310720
    float* sn      = ws + 4280320;    // [B,5,D]      20480
    float* qkvb    = ws + 4300800;    // [B,5,3D]     61440
    float* attn_o  = ws + 4362240;    // [B,5,D]      20480
    float* projb   = ws + 4382720;    // [B,5,D]      20480
    float* h1      = ws + 4403200;    // [B,5,D]      20480
    float* mlp1    = ws + 4423680;    // [B,5,4D]     81920
    float* mlp2    = ws + 4505600;    // [B,5,D]      20480
    float* se      = ws + 4526080;    // [B,5,D]      20480
    float* ag      = ws + 4546560;    // [B,5,D]      20480
    float* bg      = ws + 4567040;    // [B,5,D]      20480
    float* tmp_g   = ws + 4587520;    // [B,5,D]      20480
    float* g_out   = ws + 4608000;    // [B,5,D]      20480
    (void)in_sizes; (void)n_in; (void)out_size; (void)ws_size;

    // 1) a = X^T @ W1[:D], bp = X^T @ W1[D:], batched over (band,b)
    //    A = feature (implicit transpose: rsA=1, csA=N)
    launch_gemm(stream, feature, l_sim_w1, nullptr, a_buf, NN, DD, DD,
                1, NN, DD, 1, DD, 1,
                (long)DD*NN, (long)NBANDS*DD*NN, (long)2*DD*DD, 0,
                (long)BB*NN*DD, (long)NN*DD, 0, BB, NBANDS, 0);
    launch_gemm(stream, feature, l_sim_w1 + DD*DD, nullptr, bp_buf, NN, DD, DD,
                1, NN, DD, 1, DD, 1,
                (long)DD*NN, (long)NBANDS*DD*NN, (long)2*DD*DD, 0,
                (long)BB*NN*DD, (long)NN*DD, 0, BB, NBANDS, 0);

    // 2) static + pos_factor*pos softmaxes (per band)
    hipLaunchKernelGGL(static_pos_kernel, dim3(NBANDS*NN), dim3(NN), 0, stream,
                       l_adj_w, l_pos, l_pf, spbuf);

    // 3) fused edge-MLP + softmax + degree norm -> local adjacency
    hipLaunchKernelGGL(local_dyn_adj_kernel, dim3(NBANDS*BB*NN), dim3(256), 0, stream,
                       a_buf, bp_buf, l_sim_b1, l_sim_w2, l_sim_b2, spbuf, adj_buf);

    // 4) GCN part 1: tmp = adj @ X^T   (B operand = feature, rsB==1 -> b64 path)
    launch_gemm(stream, adj_buf, feature, nullptr, tmp_gcn, NN, DD, NN,
                NN, 1, 1, NN, DD, 1,
                (long)BB*NN*NN, (long)NN*NN, (long)DD*NN, (long)NBANDS*DD*NN,
                (long)BB*NN*DD, (long)NN*DD, 0, BB, NBANDS, 0);

    // 5) GCN part 2: local_out = tmp @ W + b, stored transposed into out3[B,5,D,N]
    launch_gemm(stream, tmp_gcn, l_gcn_w, l_gcn_b, out3, NN, DD, DD,
                DD, 1, DD, 1, 1, NN,
                (long)BB*NN*DD, (long)NN*DD, (long)DD*DD, 0,
                (long)DD*NN, (long)NBANDS*DD*NN, DD, BB, NBANDS, 1);

    // 6) scale_nodes = mean over N
    hipLaunchKernelGGL(mean_over_n_kernel, dim3(80), dim3(256), 0, stream, out3, sn);

    // 7) QKV projection: [80,256] x [256,768] + bias
    launch_gemm(stream, sn, qkv_w, qkv_b, qkvb, BB*NBANDS, 3*DD, DD,
                DD, 1, 3*DD, 1, 3*DD, 1, 0,0,0,0,0,0,0, 1, 1, 1);

    // 8) attention core (S=5, 4 heads)
    hipLaunchKernelGGL(attn_core_kernel, dim3(BB), dim3(128), 0, stream, qkvb, attn_o);

    // 9) output projection
    launch_gemm(stream, attn_o, attn_ow, attn_ob, projb, BB*NBANDS, DD, DD,
                DD, 1, DD, 1, DD, 1, 0,0,0,0,0,0,0, 1, 1, 1);

    // 10) h1 = LN(sn + proj)
    hipLaunchKernelGGL(ln_residual_kernel, dim3(BB*NBANDS), dim3(256), 0, stream,
                       sn, projb, ln1_g, ln1_b, h1);

    // 11) MLP: gelu(h1 @ W1 + b1) @ W2 + b2
    launch_gemm(stream, h1, mlp_w1, mlp_b1, mlp1, BB*NBANDS, 4*DD, DD,
                DD, 1, 4*DD, 1, 4*DD, 1, 0,0,0,0,0,0,0, 1, 1, 2);
    launch_gemm(stream, mlp1, mlp_w2, mlp_b2, mlp2, BB*NBANDS, DD, 4*DD,
                4*DD, 1, DD, 1, DD, 1, 0,0,0,0,0,0,0, 1, 1, 1);

    // 12) se = LN(h1 + mlp2)
    hipLaunchKernelGGL(ln_residual_kernel, dim3(BB*NBANDS), dim3(256), 0, stream,
                       h1, mlp2, ln2_g, ln2_b, se);

    // 13) global edge features: ag = se @ gW1[:D], bg = se @ gW1[D:]
    launch_gemm(stream, se, g_sim_w1, nullptr, ag, BB*NBANDS, DD, DD,
                DD, 1, DD, 1, DD, 1, 0,0,0,0,0,0,0, 1, 1, 0);
    launch_gemm(stream, se, g_sim_w1 + DD*DD, nullptr, bg, BB*NBANDS, DD, DD,
                DD, 1, DD, 1, DD, 1, 0,0,0,0,0,0,0, 1, 1, 0);

    // 14) global adjacency -> g_adj output
    hipLaunchKernelGGL(g_dyn_adj_kernel, dim3(BB*NBANDS), dim3(256), 0, stream,
                       ag, bg, g_sim_b1, g_sim_w2, g_sim_b2, g_adj_w, g_pos, g_pf,
                       gadj_out);

    // 15) tmp_g = g_adj @ se (K=5, plain VALU), then g_out = tmp_g @ W + b (WMMA)
    hipLaunchKernelGGL(global_agg_kernel, dim3(80), dim3(256), 0, stream,
                       gadj_out, se, tmp_g);
    launch_gemm(stream, tmp_g, g_gcn_w, g_gcn_b, g_out, BB*NBANDS, DD, DD,
                DD, 1, DD, 1, DD, 1, 0,0,0,0,0,0,0, 1, 1, 1);

    // 16) final: out[b,band,d] = g_out * (local_out . imp_w) + imp_b
    hipLaunchKernelGGL(final_proj_kernel, dim3(80), dim3(256), 0, stream,
                       out3, g_out, imp_w, imp_b, out1);
}